// SLGCNN_82076825026669
// MI455X (gfx1250) — compile-verified
//
#include <hip/hip_runtime.h>

// ---------------------------------------------------------------------------
// DGCNN head on gfx1250 (MI455X), restructured:
//   EdgeConv(x) = lrelu(bn(max_k [p(idx_k)] - p(n)))   with p = W·x  (exact:
//   conv is linear; bn scale > 0 and lrelu monotone so max commutes).
// GEMMs run on V_WMMA_F32_16X16X4_F32 (fp32 in / fp32 acc = reference
// numerics); gather/max/BN/LReLU fused in a vectorized per-point epilogue.
// ---------------------------------------------------------------------------

typedef __attribute__((ext_vector_type(2))) float v2f;
typedef __attribute__((ext_vector_type(8))) float v8f;

#define BN_EPS   1e-5f
#define NSLOPE   0.2f
#define NPTS     4096
#define NBATCH   8
#define KNN      20
#define NROWS    (NBATCH * NPTS)   // 32768 total points

__device__ __forceinline__ float lrelu(float x) { return x >= 0.f ? x : NSLOPE * x; }

template <int VEC> struct VecT;
template <> struct VecT<2> { using T = float2; };
template <> struct VecT<4> { using T = float4; };

// ---------------------------------------------------------------------------
// Layer 1 pointwise conv, C=3 (too small for WMMA): P[p][o] = sum_c w[o][c]*x[b][c][n]
// x layout is (B, 3, N) channel-major.
// ---------------------------------------------------------------------------
__global__ __launch_bounds__(256) void pconv3_kernel(const float* __restrict__ x,
                                                     const float* __restrict__ w,
                                                     float* __restrict__ P) {
    int i = blockIdx.x * 256 + threadIdx.x;   // i = p*64 + o
    int o = i & 63;
    int p = i >> 6;
    int b = p >> 12, n = p & (NPTS - 1);
    const float* xb = x + (size_t)b * 3 * NPTS + n;
    float x0 = xb[0], x1 = xb[NPTS], x2 = xb[2 * NPTS];
    const float* wr = w + o * 3;
    P[(size_t)p * 64 + o] = wr[0] * x0 + wr[1] * x1 + wr[2] * x2;
}

// ---------------------------------------------------------------------------
// Pointwise GEMM on the fp32 WMMA pipe.
//   Out[m][o] = sum_k Xin[m][inoff+k] * W[o][k]     (m = global point row)
// One wave per 16(points) x 16(outCh) tile; K stepped by 4 per WMMA.
// Fragment layouts per CDNA5 ISA 7.12.2:
//   A 16x4 f32 : lane<16 -> row=lane,K={k0,k0+1}; lane>=16 -> K={k0+2,k0+3}
//   B 4x16 f32 : same with col=lane&15
//   C/D        : VGPR j -> row = j + 8*(lane>=16), col = lane&15
// FUSED: apply BN (precomputed scale/shift per o) + LeakyReLU on store.
// ---------------------------------------------------------------------------
template <int KDIM, bool FUSED>
__global__ __launch_bounds__(256) void gemm_wmma_kernel(
    const float* __restrict__ Xin, int lda, int inoff,
    const float* __restrict__ W,
    float* __restrict__ Out, int ldo,
    const float* __restrict__ g, const float* __restrict__ bb,
    const float* __restrict__ mm, const float* __restrict__ vv) {
    int lane = threadIdx.x & 31;
    int half = lane >> 4;       // selects K pair
    int lr   = lane & 15;
    int mt   = blockIdx.x * 8 + (threadIdx.x >> 5);
    int m0   = mt * 16;
    int o0   = blockIdx.y * 16;

    const float* arow = Xin + (size_t)(m0 + lr) * lda + inoff + 2 * half;
    const float* brow = W   + (size_t)(o0 + lr) * KDIM + 2 * half;

    v8f acc = {};
#pragma unroll 8
    for (int k0 = 0; k0 < KDIM; k0 += 4) {
        v2f a = *reinterpret_cast<const v2f*>(arow + k0);   // global_load_b64
        v2f b = *reinterpret_cast<const v2f*>(brow + k0);
        acc = __builtin_amdgcn_wmma_f32_16x16x4_f32(
            /*neg_a=*/false, a, /*neg_b=*/false, b,
            /*c_mod=*/(short)0, acc, /*reuse_a=*/false, /*reuse_b=*/false);
    }

    float sc = 1.f, sh = 0.f;
    if (FUSED) {
        int o = o0 + lr;
        sc = g[o] * rsqrtf(vv[o] + BN_EPS);
        sh = bb[o] - mm[o] * sc;
    }
    float* orow = Out + (size_t)(m0 + 8 * half) * ldo + o0 + lr;
#pragma unroll
    for (int j = 0; j < 8; ++j) {
        float v = acc[j];
        if (FUSED) v = lrelu(v * sc + sh);
        orow[(size_t)j * ldo] = v;     // 16-lane contiguous per row segment
    }
}

// ---------------------------------------------------------------------------
// EdgeConv epilogue: per point, max over 20 gathered rows of P, subtract self,
// BN + LReLU, write into the concat buffer slice. One wave per point.
// Vectorized: lane l owns channels [VEC*l, VEC*l+VEC) per 32*VEC-channel group
// -> b128/b64 loads, fully coalesced, mostly L2 hits. Neighbor indices are
// loaded once (lanes 0..19) and broadcast with shuffles.
// ---------------------------------------------------------------------------
template <int O>
__global__ __launch_bounds__(256) void edge_epilogue_kernel(
    const float* __restrict__ P, const int* __restrict__ idx,
    const float* __restrict__ g, const float* __restrict__ bb,
    const float* __restrict__ mm, const float* __restrict__ vv,
    float* __restrict__ Xcat, int coff) {
    constexpr int VEC = (O >= 128) ? 4 : 2;      // floats per lane per group
    constexpr int NG  = O / (32 * VEC);          // channel groups
    using VT = typename VecT<VEC>::T;

    int lane = threadIdx.x & 31;
    int p    = blockIdx.x * 8 + (threadIdx.x >> 5);
    int b    = p >> 12;
    const int* ip = idx + (size_t)p * KNN;
    int rl = ip[lane < KNN ? lane : 0];          // indices live in lanes 0..19

    float mx[NG * VEC];
#pragma unroll
    for (int j = 0; j < NG * VEC; ++j) mx[j] = -3.4e38f;

    int base = b * NPTS;
#pragma unroll 4
    for (int k = 0; k < KNN; ++k) {
        int r = __shfl(rl, k, 32);
        const float* pr = P + (size_t)(base + r) * O;
#pragma unroll
        for (int q = 0; q < NG; ++q) {
            alignas(16) float t[VEC];
            *reinterpret_cast<VT*>(t) =
                *reinterpret_cast<const VT*>(pr + q * 32 * VEC + VEC * lane);
#pragma unroll
            for (int j = 0; j < VEC; ++j)
                mx[q * VEC + j] = fmaxf(mx[q * VEC + j], t[j]);
        }
    }

    const float* ps = P + (size_t)p * O;
    float* out = Xcat + (size_t)p * 512 + coff;
#pragma unroll
    for (int q = 0; q < NG; ++q) {
        int c0 = q * 32 * VEC + VEC * lane;
        alignas(16) float sv[VEC], gv[VEC], bv[VEC], mv[VEC], vvv[VEC], ov[VEC];
        *reinterpret_cast<VT*>(sv)  = *reinterpret_cast<const VT*>(ps + c0);
        *reinterpret_cast<VT*>(gv)  = *reinterpret_cast<const VT*>(g  + c0);
        *reinterpret_cast<VT*>(bv)  = *reinterpret_cast<const VT*>(bb + c0);
        *reinterpret_cast<VT*>(mv)  = *reinterpret_cast<const VT*>(mm + c0);
        *reinterpret_cast<VT*>(vvv) = *reinterpret_cast<const VT*>(vv + c0);
#pragma unroll
        for (int j = 0; j < VEC; ++j) {
            float sc = gv[j] * rsqrtf(vvv[j] + BN_EPS);
            float y  = (mx[q * VEC + j] - sv[j] - mv[j]) * sc + bv[j];
            ov[j] = lrelu(y);
        }
        *reinterpret_cast<VT*>(out + c0) = *reinterpret_cast<VT*>(ov);
    }
}

// ---------------------------------------------------------------------------
// Head: out[p] = lrelu(dot(H6[p][0:128], w7)). Wave per point, shfl reduce.
// ---------------------------------------------------------------------------
__global__ __launch_bounds__(256) void conv7_kernel(const float* __restrict__ H,
                                                    const float* __restrict__ w,
                                                    float* __restrict__ out) {
    int lane = threadIdx.x & 31;
    int p    = blockIdx.x * 8 + (threadIdx.x >> 5);
    const float4 h = *reinterpret_cast<const float4*>(H + (size_t)p * 128 + lane * 4);
    const float4 c = *reinterpret_cast<const float4*>(w + lane * 4);
    float s = h.x * c.x + h.y * c.y + h.z * c.z + h.w * c.w;
#pragma unroll
    for (int off = 16; off > 0; off >>= 1) s += __shfl_xor(s, off, 32);
    if (lane == 0) out[p] = lrelu(s);
}

// ---------------------------------------------------------------------------
extern "C" void kernel_launch(void* const* d_in, const int* in_sizes, int n_in,
                              void* d_out, int out_size, void* d_ws, size_t ws_size,
                              hipStream_t stream) {
    const float* x   = (const float*)d_in[0];
    const int*   idx = (const int*)d_in[1];
    const float* w[8];
    for (int i = 1; i <= 7; ++i) w[i] = (const float*)d_in[1 + i];   // w1..w7
    const float* bn[6][4];                                           // g,b,m,v per layer
    for (int i = 0; i < 6; ++i)
        for (int j = 0; j < 4; ++j) bn[i][j] = (const float*)d_in[9 + i * 4 + j];

    // Workspace: Xcat[32768][512] (concat of x1..x4, built in place) + P[32768][256]
    float* Xcat = (float*)d_ws;
    float* P    = (float*)d_ws + (size_t)NROWS * 512;

    dim3 blk(256);
    const int EPI_GRID  = NROWS / 8;           // wave per point
    const int GEMM_MB   = NROWS / (16 * 8);    // 8 m-tiles per block

    // --- EdgeConv 1 (C=3 -> 64) ---
    pconv3_kernel<<<(NROWS * 64) / 256, blk, 0, stream>>>(x, w[1], P);
    edge_epilogue_kernel<64><<<EPI_GRID, blk, 0, stream>>>(
        P, idx, bn[0][0], bn[0][1], bn[0][2], bn[0][3], Xcat, 0);

    // --- EdgeConv 2 (64 -> 64) ---
    gemm_wmma_kernel<64, false><<<dim3(GEMM_MB, 64 / 16), blk, 0, stream>>>(
        Xcat, 512, 0, w[2], P, 64, nullptr, nullptr, nullptr, nullptr);
    edge_epilogue_kernel<64><<<EPI_GRID, blk, 0, stream>>>(
        P, idx, bn[1][0], bn[1][1], bn[1][2], bn[1][3], Xcat, 64);

    // --- EdgeConv 3 (64 -> 128) ---
    gemm_wmma_kernel<64, false><<<dim3(GEMM_MB, 128 / 16), blk, 0, stream>>>(
        Xcat, 512, 64, w[3], P, 128, nullptr, nullptr, nullptr, nullptr);
    edge_epilogue_kernel<128><<<EPI_GRID, blk, 0, stream>>>(
        P, idx, bn[2][0], bn[2][1], bn[2][2], bn[2][3], Xcat, 128);

    // --- EdgeConv 4 (128 -> 256) ---
    gemm_wmma_kernel<128, false><<<dim3(GEMM_MB, 256 / 16), blk, 0, stream>>>(
        Xcat, 512, 128, w[4], P, 256, nullptr, nullptr, nullptr, nullptr);
    edge_epilogue_kernel<256><<<EPI_GRID, blk, 0, stream>>>(
        P, idx, bn[3][0], bn[3][1], bn[3][2], bn[3][3], Xcat, 256);

    // --- PointConv 5 (512 -> 256), fused BN+LReLU, H5 lives in P buffer ---
    gemm_wmma_kernel<512, true><<<dim3(GEMM_MB, 256 / 16), blk, 0, stream>>>(
        Xcat, 512, 0, w[5], P, 256, bn[4][0], bn[4][1], bn[4][2], bn[4][3]);

    // --- PointConv 6 (256 -> 128), fused, H6 overwrites Xcat region ---
    gemm_wmma_kernel<256, true><<<dim3(GEMM_MB, 128 / 16), blk, 0, stream>>>(
        P, 256, 0, w[6], Xcat, 128, bn[5][0], bn[5][1], bn[5][2], bn[5][3]);

    // --- Head (128 -> 1) + LReLU; output layout (B, N, 1) == flat p index ---
    conv7_kernel<<<EPI_GRID, blk, 0, stream>>>(Xcat, w[7], (float*)d_out);
}